// QuantumLayer_65481071400024
// MI455X (gfx1250) — compile-verified
//
#include <hip/hip_runtime.h>
#include <hip/hip_bf16.h>

// QuantumLayer closed form:
//   out[n, i] = prod_{j<=i} cos(x[n, j])
// Derivation: RX product state -> CNOT chain is a basis-state permutation with
// final bit i = XOR of initial bits 0..i, so <Z_i> = prod_{j<=i} E[(-1)^{b_j}]
//            = prod_{j<=i} cos(x_j). RZ(weights) is diagonal in the Z basis,
// so |amp|^2 (and hence every <Z_i>) is independent of `weights`.
//
// cos via the hardware transcendental: v_cos_f32 takes its argument in
// revolutions (units of 2*pi). For |x| <~ 6 (inputs ~ N(0,1)) the scaling
// multiply adds phase error <= |x|*2^-24 rad => output error ~1e-6: exact
// enough for f32 validation, and 2 VALU ops instead of ~80 for OCML cosf.
//
// The 16(tokens) x 12 result tile is routed through the CDNA5 matrix unit as
// D = M x I via 3 accumulating V_WMMA_F32_16X16X4_F32 ops (B_c = identity
// row-blocks for K=0..11). Exact in FP (1.0*m + 0*finite); D columns 12..15
// are zero and never stored. Stores use the documented 16x16 f32 C/D layout.

typedef __attribute__((ext_vector_type(2))) float v2f;
typedef __attribute__((ext_vector_type(8))) float v8f;

__device__ __forceinline__ float fast_cos(float x) {
  // 1/(2*pi); v_cos_f32 input is in revolutions.
  return __builtin_amdgcn_cosf(x * 0.15915494309189535f);
}

__global__ __launch_bounds__(256) void quantum_layer_kernel(
    const float* __restrict__ x, float* __restrict__ out, int n_tokens) {
  const int lane = threadIdx.x & 31;          // wave32 lane
  const int wave = (blockIdx.x * blockDim.x + threadIdx.x) >> 5;
  const int token_base = wave * 16;           // 16 tokens per wave tile
  if (token_base >= n_tokens) return;         // wave-uniform exit

  const int  t  = lane & 15;                  // tile row / output column
  const bool hi = (lane >= 16);               // lane half selects K pair

  // Clamp so partial tiles never read OOB (EXEC stays all-1s for WMMA).
  long token = token_base + t;
  if (token >= n_tokens) token = n_tokens - 1;

  // 12 floats per token = 48 B stride -> always 16 B aligned: 3x b128 loads.
  const float4* __restrict__ xp = (const float4*)(x + token * 12);
  const float4 q0 = xp[0];
  const float4 q1 = xp[1];
  const float4 q2 = xp[2];

  // Prefix products of cos over the 12 wires, as named scalars (no local
  // array -> no dynamic-index select trees in codegen).
  const float p0  = fast_cos(q0.x);
  const float p1  = p0  * fast_cos(q0.y);
  const float p2  = p1  * fast_cos(q0.z);
  const float p3  = p2  * fast_cos(q0.w);
  const float p4  = p3  * fast_cos(q1.x);
  const float p5  = p4  * fast_cos(q1.y);
  const float p6  = p5  * fast_cos(q1.z);
  const float p7  = p6  * fast_cos(q1.w);
  const float p8  = p7  * fast_cos(q2.x);
  const float p9  = p8  * fast_cos(q2.y);
  const float p10 = p9  * fast_cos(q2.z);
  const float p11 = p10 * fast_cos(q2.w);

  // D = sum_{c=0..2} A_c (16x4 slice of M) * B_c (4x16 identity row-block).
  // f32 A 16x4 layout: lanes 0-15 hold {K=4c+0,4c+1}, lanes 16-31 {4c+2,4c+3}.
  // f32 B 4x16 layout: lane's col N=t holds the same K pair.
  const int kbase = hi ? 2 : 0;               // one v_cndmask
  v8f acc = {};
  {
    v2f a, b;
    a.x = hi ? p2 : p0;
    a.y = hi ? p3 : p1;
    b.x = (t == kbase + 0) ? 1.0f : 0.0f;
    b.y = (t == kbase + 1) ? 1.0f : 0.0f;
    acc = __builtin_amdgcn_wmma_f32_16x16x4_f32(false, a, false, b,
                                                (short)0, acc, false, false);
  }
  {
    v2f a, b;
    a.x = hi ? p6 : p4;
    a.y = hi ? p7 : p5;
    b.x = (t == kbase + 4) ? 1.0f : 0.0f;
    b.y = (t == kbase + 5) ? 1.0f : 0.0f;
    acc = __builtin_amdgcn_wmma_f32_16x16x4_f32(false, a, false, b,
                                                (short)0, acc, false, false);
  }
  {
    v2f a, b;
    a.x = hi ? p10 : p8;
    a.y = hi ? p11 : p9;
    b.x = (t == kbase + 8) ? 1.0f : 0.0f;
    b.y = (t == kbase + 9) ? 1.0f : 0.0f;
    acc = __builtin_amdgcn_wmma_f32_16x16x4_f32(false, a, false, b,
                                                (short)0, acc, false, false);
  }

  // 16x16 f32 C/D layout: VGPR r -> row r (lanes 0-15, N=lane) / row r+8
  // (lanes 16-31, N=lane-16). Columns t < 12 are real wires; store those.
  const bool full = (token_base + 16 <= n_tokens);  // wave-uniform
  if (t < 12) {
    const int row0 = token_base + (hi ? 8 : 0);
    float* __restrict__ op = out + (long)row0 * 12 + t;
    if (full) {
      // Fast path: 8 unguarded dword stores, immediate offsets off one base.
#pragma unroll
      for (int r = 0; r < 8; ++r) op[r * 12] = acc[r];
    } else {
#pragma unroll
      for (int r = 0; r < 8; ++r)
        if (row0 + r < n_tokens) op[r * 12] = acc[r];
    }
  }
}

extern "C" void kernel_launch(void* const* d_in, const int* in_sizes, int n_in,
                              void* d_out, int out_size, void* d_ws, size_t ws_size,
                              hipStream_t stream) {
  const float* x = (const float*)d_in[0];
  // d_in[1] (weights, 12 floats) is provably irrelevant: RZ is diagonal in the
  // Z measurement basis, so it cannot change any <Z_i>.
  float* out = (float*)d_out;

  const int n_tokens = in_sizes[0] / 12;            // 16*512 = 8192
  const int waves    = (n_tokens + 15) / 16;        // 16 tokens per wave
  const int blocks   = (waves + 7) / 8;             // 8 waves (256 thr) / block

  hipLaunchKernelGGL(quantum_layer_kernel, dim3(blocks), dim3(256), 0, stream,
                     x, out, n_tokens);
}